// GraphEncoder_52218212384876
// MI455X (gfx1250) — compile-verified
//
#include <hip/hip_runtime.h>
#include <hip/hip_bf16.h>

// Problem constants (match reference)
#define BB 16
#define NN 2048
#define EE 32768
#define DD 256
#define RR 8
#define LL 2

typedef __attribute__((ext_vector_type(2))) float v2f;
typedef __attribute__((ext_vector_type(8))) float v8f;

// ---------------------------------------------------------------------------
// Kernel 1: x0[b,n,:] = emb[nodes[b,n], :]   (float4 vectorized gather)
// grid covers exactly B*N*(D/4) threads
// ---------------------------------------------------------------------------
__global__ __launch_bounds__(256) void gather_embed(const int* __restrict__ nodes,
                                                    const float* __restrict__ emb,
                                                    float* __restrict__ x0) {
    size_t idx = (size_t)blockIdx.x * 256 + threadIdx.x;   // [0, B*N*D/4)
    int d4 = (int)(idx & (DD / 4 - 1));                    // D/4 == 64
    size_t bn = idx >> 6;
    int node = nodes[bn];
    const float4* __restrict__ e4 = (const float4*)emb;
    ((float4*)x0)[idx] = e4[(size_t)node * (DD / 4) + d4];
}

// ---------------------------------------------------------------------------
// Kernel 2: deg[b, dst] += 1  (degrees are layer-invariant; run once)
// ---------------------------------------------------------------------------
__global__ __launch_bounds__(256) void count_deg(const int* __restrict__ edges,
                                                 float* __restrict__ deg) {
    size_t idx = (size_t)blockIdx.x * 256 + threadIdx.x;   // [0, B*E)
    int b = (int)(idx >> 15);                              // E == 32768
    int e = (int)(idx & (EE - 1));
    int dst = edges[(size_t)b * 2 * EE + EE + e];
    atomicAdd(deg + (size_t)b * NN + dst, 1.0f);
}

// ---------------------------------------------------------------------------
// Kernel 3: S[b, type_e, dst_e, :] += x[b, src_e, :]
// One wave (32 lanes) per edge; lane l handles 8 consecutive floats.
// ---------------------------------------------------------------------------
__global__ __launch_bounds__(256) void scatter_edges(const float* __restrict__ x,
                                                     const int* __restrict__ edges,
                                                     const int* __restrict__ types,
                                                     float* __restrict__ S) {
    int lane = threadIdx.x & 31;
    int ge = blockIdx.x * 8 + (threadIdx.x >> 5);          // global edge id, [0, B*E)
    int b = ge >> 15;                                      // E == 32768
    int e = ge & (EE - 1);
    const int* eb = edges + (size_t)b * 2 * EE;
    int src = eb[e];
    int dst = eb[EE + e];
    int t = types[(size_t)b * EE + e];

    const float4* __restrict__ xp =
        (const float4*)(x + ((size_t)b * NN + src) * DD) + lane * 2;
    float4 v0 = xp[0];
    float4 v1 = xp[1];

    float* sp = S + ((((size_t)b * RR + t) * NN + dst) * DD) + lane * 8;
    atomicAdd(sp + 0, v0.x);
    atomicAdd(sp + 1, v0.y);
    atomicAdd(sp + 2, v0.z);
    atomicAdd(sp + 3, v0.w);
    atomicAdd(sp + 4, v1.x);
    atomicAdd(sp + 5, v1.y);
    atomicAdd(sp + 6, v1.z);
    atomicAdd(sp + 7, v1.w);
}

// ---------------------------------------------------------------------------
// Kernel 4: per 16x16 output tile (one wave each):
//   accS = sum_r S[b,r] @ W[r]          (8 GEMM passes, K=256)
//   accR = x @ W_root                   (1 GEMM pass)
//   out  = accS / max(deg,1) + accR + bias   (+ ReLU for layer 0)
// Uses V_WMMA_F32_16X16X4_F32 with the ISA 16x4-A / 4x16-B lane layouts:
//   A: lanes 0-15 -> M=lane, K=k0..k0+1 ; lanes 16-31 -> M=lane-16, K=k0+2..k0+3
//   B: lanes 0-15 -> N=lane, K=k0..k0+1 ; lanes 16-31 -> N=lane-16, K=k0+2..k0+3
//   C/D: VGPR i holds row M=i (lanes 0-15) / M=8+i (lanes 16-31), col N=lane%16
// ---------------------------------------------------------------------------
__global__ __launch_bounds__(256) void rgcn_wmma(const float* __restrict__ xin,
                                                 const float* __restrict__ S,
                                                 const float* __restrict__ deg,
                                                 const float* __restrict__ Wl,
                                                 const float* __restrict__ Wroot,
                                                 const float* __restrict__ bias,
                                                 float* __restrict__ xout,
                                                 int do_relu) {
    const int MT = NN / 16;   // 128 row tiles per batch
    const int NT = DD / 16;   // 16 col tiles

    int lane = threadIdx.x & 31;
    int wid = blockIdx.x * 8 + (threadIdx.x >> 5);         // [0, B*MT*NT)
    int b = wid / (MT * NT);
    int rem = wid - b * (MT * NT);
    int mt = rem / NT;
    int nt = rem - mt * NT;

    int mrow = lane & 15;     // row within tile for A loads
    int hi = lane >> 4;       // 0: K pair {0,1}, 1: K pair {2,3}
    int col = nt * 16 + (lane & 15);

    // ---- relation GEMMs: accS = sum_r S[b,r,tile] @ W[r] ----
    v8f accS = {0.f, 0.f, 0.f, 0.f, 0.f, 0.f, 0.f, 0.f};
    for (int r = 0; r < RR; ++r) {
        const float* __restrict__ A =
            S + ((((size_t)b * RR + r) * NN) + (size_t)mt * 16 + mrow) * DD + hi * 2;
        const float* __restrict__ Bm = Wl + (size_t)r * DD * DD;
#pragma unroll 4
        for (int k0 = 0; k0 < DD; k0 += 4) {
            v2f a = *(const v2f*)(A + k0);
            v2f bvec;
            bvec.x = Bm[(size_t)(k0 + hi * 2) * DD + col];
            bvec.y = Bm[(size_t)(k0 + hi * 2 + 1) * DD + col];
            accS = __builtin_amdgcn_wmma_f32_16x16x4_f32(
                false, a, false, bvec, (short)0, accS, false, false);
        }
    }

    // ---- root GEMM: accR = xin[tile] @ W_root ----
    v8f accR = {0.f, 0.f, 0.f, 0.f, 0.f, 0.f, 0.f, 0.f};
    {
        const float* __restrict__ A =
            xin + ((size_t)b * NN + (size_t)mt * 16 + mrow) * DD + hi * 2;
#pragma unroll 4
        for (int k0 = 0; k0 < DD; k0 += 4) {
            v2f a = *(const v2f*)(A + k0);
            v2f bvec;
            bvec.x = Wroot[(size_t)(k0 + hi * 2) * DD + col];
            bvec.y = Wroot[(size_t)(k0 + hi * 2 + 1) * DD + col];
            accR = __builtin_amdgcn_wmma_f32_16x16x4_f32(
                false, a, false, bvec, (short)0, accR, false, false);
        }
    }

    // ---- epilogue: normalize message sum, add root + bias, ReLU ----
    float bv = bias[col];
    const float* dg = deg + (size_t)b * NN + (size_t)mt * 16 + hi * 8;
    float* out = xout + ((size_t)b * NN + (size_t)mt * 16 + hi * 8) * DD + col;
#pragma unroll
    for (int i = 0; i < 8; ++i) {
        float dv = dg[i];
        dv = dv > 1.0f ? dv : 1.0f;
        float v = accS[i] / dv + accR[i] + bv;
        if (do_relu) v = v > 0.0f ? v : 0.0f;
        out[(size_t)i * DD] = v;
    }
}

// ---------------------------------------------------------------------------
// Host launch
// ---------------------------------------------------------------------------
extern "C" void kernel_launch(void* const* d_in, const int* in_sizes, int n_in,
                              void* d_out, int out_size, void* d_ws, size_t ws_size,
                              hipStream_t stream) {
    const int* nodes = (const int*)d_in[0];     // (B,N)
    const int* edges = (const int*)d_in[1];     // (B,2,E)
    const int* types = (const int*)d_in[2];     // (B,E)
    const float* emb = (const float*)d_in[3];   // (V,D)
    const float* W = (const float*)d_in[4];     // (L,R,D,D)
    const float* Wroot = (const float*)d_in[5]; // (L,D,D)
    const float* bias = (const float*)d_in[6];  // (L,D)

    const size_t xElems = (size_t)BB * NN * DD;           // 8.4M floats
    const size_t sElems = (size_t)BB * RR * NN * DD;      // 67.1M floats

    float* x0 = (float*)d_ws;
    float* x1 = x0 + xElems;
    float* S = x1 + xElems;
    float* deg = S + sElems;

    // 1) embedding gather
    {
        size_t threads = xElems / 4;
        gather_embed<<<(unsigned)(threads / 256), 256, 0, stream>>>(nodes, emb, x0);
    }

    // 2) degrees (layer-invariant)
    hipMemsetAsync(deg, 0, (size_t)BB * NN * sizeof(float), stream);
    count_deg<<<(BB * EE) / 256, 256, 0, stream>>>(edges, deg);

    // 3) two RGCN layers
    for (int l = 0; l < LL; ++l) {
        const float* xin = (l == 0) ? x0 : x1;
        float* xout = (l == 0) ? x1 : (float*)d_out;

        hipMemsetAsync(S, 0, sElems * sizeof(float), stream);
        scatter_edges<<<(BB * EE) / 8, 256, 0, stream>>>(xin, edges, types, S);

        int tiles = BB * (NN / 16) * (DD / 16);           // 32768 waves
        rgcn_wmma<<<tiles / 8, 256, 0, stream>>>(
            xin, S, deg,
            W + (size_t)l * RR * DD * DD,
            Wroot + (size_t)l * DD * DD,
            bias + (size_t)l * DD,
            xout, (l < LL - 1) ? 1 : 0);
    }
}